// ConvLayer_67585605370034
// MI455X (gfx1250) — compile-verified
//
#include <hip/hip_runtime.h>
#include <hip/hip_bf16.h>
#include <math.h>

typedef __attribute__((ext_vector_type(2))) float v2f;
typedef __attribute__((ext_vector_type(8))) float v8f;

#define NUM_SEG 4096

// ---------------------------------------------------------------------------
// Kernel A: h = x @ W_conv + b_conv  (WMMA f32 16x16x4), att = h @ W_att + b_att
// One block = 32 rows. 8 waves; each wave owns a 16-col tile and TWO 16-row
// M-tiles (acc0/acc1), so every B fragment (one ds_load_b64 from the packed
// W staged in LDS) feeds two WMMAs. Inner K-loop is pure LDS + WMMA.
// ---------------------------------------------------------------------------
__global__ __launch_bounds__(256) void gemm_att_kernel(
    const float* __restrict__ x, const float* __restrict__ Wc,
    const float* __restrict__ bc, const float* __restrict__ Wa,
    const float* __restrict__ ba, float* __restrict__ h,
    float* __restrict__ att)
{
    __shared__ float  xs[32 * 128];    // 16 KB x row-tile
    __shared__ float2 wp[64 * 128];    // 64 KB W packed as K-pairs: wp[p*128+n] = {W[2p][n], W[2p+1][n]}
    __shared__ float  latt[32];        // per-row att accumulators

    const int tid  = threadIdx.x;
    const int wave = tid >> 5;         // 0..7 -> column tile
    const int lane = tid & 31;
    const int rowbase = blockIdx.x * 32;

    // stage x tile: 4096 floats / 256 threads (coalesced)
    #pragma unroll
    for (int i = 0; i < 16; ++i)
        xs[tid + i * 256] = x[(size_t)rowbase * 128 + tid + i * 256];
    // stage W packed: 8192 float2 (both source reads coalesced across n)
    for (int i = tid; i < 64 * 128; i += 256) {
        const int p = i >> 7, nn = i & 127;
        float2 v;
        v.x = Wc[(size_t)(2 * p) * 128 + nn];
        v.y = Wc[(size_t)(2 * p + 1) * 128 + nn];
        wp[i] = v;
    }
    if (tid < 32) latt[tid] = ba[0];
    __syncthreads();

    const int nloc  = lane & 15;           // column within tile
    const int n     = wave * 16 + nloc;    // global output channel
    const int khalf = (lane >> 4) * 2;     // K sub-offset 0 or 2
    const int m     = lane & 15;           // A-row for this lane (within M-tile)

    // C/D init: bias broadcast per column (every D slot in this lane is col n)
    const float bias = bc[n];
    v8f acc0, acc1;
    #pragma unroll
    for (int v = 0; v < 8; ++v) { acc0[v] = bias; acc1[v] = bias; }

    // K loop: 64 x V_WMMA_F32_16X16X4_F32, B fragment reused by both M-tiles
    #pragma unroll
    for (int k0 = 0; k0 < 128; k0 += 4) {
        const int k = k0 + khalf;
        v2f a0, a1, b;
        a0[0] = xs[m * 128 + k];
        a0[1] = xs[m * 128 + k + 1];
        a1[0] = xs[(m + 16) * 128 + k];
        a1[1] = xs[(m + 16) * 128 + k + 1];
        const float2 bv = wp[(k >> 1) * 128 + n];   // single ds_load_b64
        b[0] = bv.x;
        b[1] = bv.y;
        acc0 = __builtin_amdgcn_wmma_f32_16x16x4_f32(
            false, a0, false, b, (short)0, acc0, false, false);
        acc1 = __builtin_amdgcn_wmma_f32_16x16x4_f32(
            false, a1, false, b, (short)0, acc1, false, false);
    }

    // Store h + reduce att per row via LDS float atomics.
    // D layout: VGPR v -> row v (lanes 0-15) / row v+8 (lanes 16-31), col = nloc.
    const float wan  = Wa[n];
    const int   mofs = (lane >> 4) * 8;
    #pragma unroll
    for (int v = 0; v < 8; ++v) {
        const int r0 = v + mofs;            // M-tile 0
        h[(size_t)(rowbase + r0) * 128 + n] = acc0[v];
        atomicAdd(&latt[r0], acc0[v] * wan);
        const int r1 = 16 + v + mofs;       // M-tile 1
        h[(size_t)(rowbase + r1) * 128 + n] = acc1[v];
        atomicAdd(&latt[r1], acc1[v] * wan);
    }
    __syncthreads();
    if (tid < 32) att[rowbase + tid] = latt[tid];
}

// ---------------------------------------------------------------------------
// Kernel B: one block per segment (segment_idx is sorted -> contiguous range).
// Segment softmax -> a, weighted moments -> ACN normalize -> GroupNorm -> ReLU.
// ---------------------------------------------------------------------------
__global__ __launch_bounds__(256) void seg_norm_kernel(
    const float* __restrict__ h, const float* __restrict__ att,
    const float* __restrict__ wpri, const int* __restrict__ seg,
    const float* __restrict__ gamma, const float* __restrict__ beta,
    float* __restrict__ out, float* __restrict__ a_out, int N)
{
    __shared__ int   bnd[2];
    __shared__ float red[16];
    __shared__ float S1[128];
    __shared__ float S2[128];
    __shared__ float stat[4];   // [0]=max  [1]=1/D  [2]=sumA

    const int s   = blockIdx.x;
    const int tid = threadIdx.x;
    const int lane = tid & 31;
    const int wave = tid >> 5;

    if (tid < 2) {
        const int key = s + tid;               // lower_bound(s), lower_bound(s+1)
        int lo = 0, hi = N;
        while (lo < hi) { int mid = (lo + hi) >> 1; if (seg[mid] < key) lo = mid + 1; else hi = mid; }
        bnd[tid] = lo;
    }
    __syncthreads();
    const int start = bnd[0], end = bnd[1];
    if (start >= end) return;                  // empty segment (uniform exit)

    // ---- pass 1: segment max of att ----
    float m = -INFINITY;
    for (int i = start + tid; i < end; i += 256) m = fmaxf(m, att[i]);
    #pragma unroll
    for (int off = 16; off > 0; off >>= 1) m = fmaxf(m, __shfl_xor(m, off));
    if (lane == 0) red[wave] = m;
    __syncthreads();
    if (tid == 0) {
        float mm = red[0];
        #pragma unroll
        for (int i = 1; i < 8; ++i) mm = fmaxf(mm, red[i]);
        stat[0] = mm;
    }
    __syncthreads();
    m = stat[0];

    // ---- pass 2: sumE = sum(e), sumEW = sum(e * weight_pri) ----
    float se = 0.f, sw = 0.f;
    for (int i = start + tid; i < end; i += 256) {
        const float e = __expf(att[i] - m);
        se += e;
        sw += e * wpri[i];
    }
    #pragma unroll
    for (int off = 16; off > 0; off >>= 1) { se += __shfl_xor(se, off); sw += __shfl_xor(sw, off); }
    if (lane == 0) { red[wave] = se; red[wave + 8] = sw; }
    __syncthreads();
    if (tid == 0) {
        float tse = 0.f, tsw = 0.f;
        #pragma unroll
        for (int i = 0; i < 8; ++i) { tse += red[i]; tsw += red[i + 8]; }
        // a_i = (e_i/sumE)*w_i / max(sumEW/sumE, 1e-3) = e_i*w_i / max(sumEW, 1e-3*sumE)
        stat[1] = 1.f / fmaxf(tsw, 1e-3f * tse);
    }
    __syncthreads();
    const float invD = stat[1];

    // ---- pass 3: a_i, sumA, weighted channel moments S1 = sum(a*h), S2 = sum(a*h^2) ----
    const int c = tid & 127;       // channel
    const int r = tid >> 7;        // row-lane 0/1
    float s1 = 0.f, s2 = 0.f, sa = 0.f;
    for (int i = start + r; i < end; i += 2) {
        const float ai = __expf(att[i] - m) * wpri[i] * invD;
        const float hv = h[(size_t)i * 128 + c];
        s1 += ai * hv;
        s2 += ai * hv * hv;
        if (c == 0) { sa += ai; a_out[i] = ai; }
    }
    if (r == 0) { S1[c] = s1; S2[c] = s2; if (c == 0) stat[2] = sa; }
    __syncthreads();
    if (r == 1) { S1[c] += s1; S2[c] += s2; if (c == 0) stat[2] += sa; }
    __syncthreads();
    if (tid < 128) {
        const float mean = S1[tid];
        const float sumA = stat[2];
        // sum(a*(h-mean)^2) = S2 - 2*mean^2 + mean^2*sumA   (mean == S1)
        const float var = S2[tid] - 2.f * mean * mean + mean * mean * sumA;
        S2[tid] = rsqrtf(var + 1e-3f);        // 1/std  (ACN_EPS)
    }
    __syncthreads();

    // ---- pass 4: ACN apply + GroupNorm(32 groups of 4 ch) + ReLU ----
    // c%4 == lane%4, so a 4-channel group = 4 consecutive lanes of one wave32.
    const float gam = gamma[c], bet = beta[c];
    for (int i = start + r; i < end; i += 2) {
        const float o = (h[(size_t)i * 128 + c] - S1[c]) * S2[c];
        float sumo = o + __shfl_xor(o, 1);  sumo += __shfl_xor(sumo, 2);
        float sq   = o * o;
        float sums = sq + __shfl_xor(sq, 1); sums += __shfl_xor(sums, 2);
        const float mu  = sumo * 0.25f;
        const float var = sums * 0.25f - mu * mu;
        const float gn  = (o - mu) * rsqrtf(var + 1e-5f) * gam + bet;
        out[(size_t)i * 128 + c] = fmaxf(gn, 0.f);
    }
}

// ---------------------------------------------------------------------------
extern "C" void kernel_launch(void* const* d_in, const int* in_sizes, int n_in,
                              void* d_out, int out_size, void* d_ws, size_t ws_size,
                              hipStream_t stream) {
    const float* x     = (const float*)d_in[0];
    const float* wpri  = (const float*)d_in[1];
    const float* Wc    = (const float*)d_in[2];
    const float* bc    = (const float*)d_in[3];
    const float* Wa    = (const float*)d_in[4];
    const float* ba    = (const float*)d_in[5];
    const float* gamma = (const float*)d_in[6];
    const float* beta  = (const float*)d_in[7];
    const int*   seg   = (const int*)d_in[8];

    const int N = in_sizes[1];               // weight_pri has N elements

    float* hbuf   = (float*)d_ws;            // [N,128]
    float* attbuf = hbuf + (size_t)N * 128;  // [N]

    float* out   = (float*)d_out;            // [N,128]
    float* a_out = out + (size_t)N * 128;    // [N]  (tuple second output)

    gemm_att_kernel<<<N / 32, 256, 0, stream>>>(x, Wc, bc, Wa, ba, hbuf, attbuf);
    seg_norm_kernel<<<NUM_SEG, 256, 0, stream>>>(hbuf, attbuf, wpri, seg,
                                                 gamma, beta, out, a_out, N);
}